// GNN_drug_18133351924123
// MI455X (gfx1250) — compile-verified
//
#include <hip/hip_runtime.h>

typedef float v2f __attribute__((ext_vector_type(2)));
typedef float v8f __attribute__((ext_vector_type(8)));

#define N_NODES  300000
#define N_EDGES  1200000
#define N_GRAPHS 10000
#define DIM      128
#define IN_DIM   77
#define KP0      80          // IN_DIM rounded up to multiple of 4
#define N_LAYERS 3
#define BN_EPS   1e-5f

// ---------------------------------------------------------------- elementwise
__global__ void gin_copy_kernel(const float* __restrict__ src,
                                float* __restrict__ dst, int n) {
    int i = blockIdx.x * blockDim.x + threadIdx.x;
    int stride = gridDim.x * blockDim.x;
    for (; i < n; i += stride) dst[i] = src[i];
}

__global__ void gin_zero_kernel(float* __restrict__ p, int n) {
    int i = blockIdx.x * blockDim.x + threadIdx.x;
    if (i < n) p[i] = 0.0f;
}

// zero-pad W1_0 [77,128] -> Wp [80,128] (rows 77..79 = 0)
__global__ void gin_pad_w_kernel(const float* __restrict__ W,
                                 float* __restrict__ Wp) {
    int i = blockIdx.x * blockDim.x + threadIdx.x;      // 80*128 elems
    if (i < KP0 * DIM) {
        int k = i >> 7;
        Wp[i] = (k < IN_DIM) ? W[i] : 0.0f;
    }
}

// ------------------------------------------------- edge scatter (segment_sum)
// t pre-initialized with h, so after scatter t = h + sum_{j in N(i)} h_j
// 128-wide layers: float4 gather + 4 float atomics per (edge, quad) task
__global__ void gin_edge_scatter128_kernel(const float* __restrict__ h,
                                           const int* __restrict__ esrc,
                                           const int* __restrict__ edst,
                                           float* __restrict__ t) {
    int i = blockIdx.x * blockDim.x + threadIdx.x;
    int stride = gridDim.x * blockDim.x;
    const int total = N_EDGES * 32;                     // 32 quads per edge
    for (; i < total; i += stride) {
        int e = i >> 5;
        int q = i & 31;
        int s = esrc[e];
        int d = edst[e];
        const float4 v = ((const float4*)(h + (size_t)s * DIM))[q];
        float* tp = t + (size_t)d * DIM + q * 4;
        atomicAdd(tp + 0, v.x);
        atomicAdd(tp + 1, v.y);
        atomicAdd(tp + 2, v.z);
        atomicAdd(tp + 3, v.w);
    }
}

// layer-0 (F = 77) scatter; F compile-time so /77 folds to magic multiply
__global__ void gin_edge_scatter77_kernel(const float* __restrict__ h,
                                          const int* __restrict__ esrc,
                                          const int* __restrict__ edst,
                                          float* __restrict__ t) {
    const int F = IN_DIM;
    int i = blockIdx.x * blockDim.x + threadIdx.x;
    int stride = gridDim.x * blockDim.x;
    const int total = N_EDGES * F;
    for (; i < total; i += stride) {
        int e = i / F;
        int f = i - e * F;
        atomicAdd(&t[edst[e] * F + f], h[esrc[e] * F + f]);
    }
}

// ---------------------------------------------------------- WMMA fp32 GEMM
// Out[N,128] = relu(A[N,K] @ Wp[KP,128] + bias)   (Wp zero-padded to KP rows)
// block = 256 thr = 8 waves; block owns 16 rows; wave w owns cols [16w,16w+16)
// Fully-unrolled V_WMMA_F32_16X16X4_F32 K-loop, no bounds checks in hot loop.
// In-place (Out == A, K == DIM) is safe: block stages its rows to LDS first.
template <int K, int KPAD>
__global__ void __launch_bounds__(256)
gin_gemm_bias_relu_wmma(const float* __restrict__ A, const float* __restrict__ Wp,
                        const float* __restrict__ bias, float* __restrict__ Out) {
    __shared__ float As[16 * KPAD];
    const int tid  = threadIdx.x;
    const int bm   = blockIdx.x;                // 16-row slab
    const int lane = tid & 31;
    const int wave = tid >> 5;                  // 0..7 -> col tile
    const int half = lane >> 4;                 // K sub-pair select
    const int l16  = lane & 15;

    // stage 16 x KPAD slab of A (zero-pad c in [K, KPAD) at compile time)
    for (int idx = tid; idx < 16 * KPAD; idx += 256) {
        const int r = idx / KPAD;
        const int c = idx - r * KPAD;
        if constexpr (K == KPAD) {
            As[idx] = A[(bm * 16 + r) * K + c];
        } else {
            As[idx] = (c < K) ? A[(bm * 16 + r) * K + c] : 0.0f;
        }
    }
    __syncthreads();

    v8f acc = {};
    const int coln = wave * 16 + l16;
#pragma unroll
    for (int k = 0; k < KPAD; k += 4) {
        const int ka = k + 2 * half;
        v2f a, b;
        // A frag: lanes 0-15 -> K=k,k+1 ; lanes 16-31 -> K=k+2,k+3 ; M=l16
        a.x = As[l16 * KPAD + ka];
        a.y = As[l16 * KPAD + ka + 1];
        // B frag (mirror of A): N=coln, same K split; Wp pre-padded, no checks
        b.x = Wp[ka * DIM + coln];
        b.y = Wp[(ka + 1) * DIM + coln];
        acc = __builtin_amdgcn_wmma_f32_16x16x4_f32(
            false, a, false, b, (short)0, acc, false, false);
    }

    const float bv = bias[coln];
    const int row_base = bm * 16 + 8 * half;    // C VGPR r -> rows r / r+8
#pragma unroll
    for (int r = 0; r < 8; ++r) {
        float v = acc[r] + bv;
        v = v > 0.0f ? v : 0.0f;
        Out[(row_base + r) * DIM + coln] = v;
    }
}

// ------------------------------------------------------------------ BatchNorm
// stats[0:128]=sum  [128:256]=sumsq  [256:384]=scale  [384:512]=shift
__global__ void gin_bn_stats_kernel(const float* __restrict__ h,
                                    float* __restrict__ stats, int rowsPerBlock) {
    int f  = threadIdx.x;                       // 0..127
    int r0 = blockIdx.x * rowsPerBlock;
    int r1 = r0 + rowsPerBlock;
    if (r1 > N_NODES) r1 = N_NODES;
    float s = 0.0f, s2 = 0.0f;
    for (int i = r0; i < r1; ++i) {
        float v = h[i * DIM + f];
        s += v;
        s2 += v * v;
    }
    atomicAdd(&stats[f], s);
    atomicAdd(&stats[DIM + f], s2);
}

__global__ void gin_bn_finalize_kernel(float* __restrict__ stats,
                                       const float* __restrict__ gamma,
                                       const float* __restrict__ beta) {
    int f = threadIdx.x;
    float inv_n = 1.0f / (float)N_NODES;
    float mean  = stats[f] * inv_n;
    float var   = stats[DIM + f] * inv_n - mean * mean;
    float sc    = gamma[f] * rsqrtf(var + BN_EPS);
    stats[2 * DIM + f] = sc;
    stats[3 * DIM + f] = beta[f] - mean * sc;
}

__global__ void gin_bn_apply_kernel(float* __restrict__ h,
                                    const float* __restrict__ stats) {
    int i = blockIdx.x * blockDim.x + threadIdx.x;
    int stride = gridDim.x * blockDim.x;
    const int total = N_NODES * DIM;
    for (; i < total; i += stride) {
        int f = i & (DIM - 1);
        h[i] = h[i] * stats[2 * DIM + f] + stats[3 * DIM + f];
    }
}

// --------------------------------------------------- per-graph segment_max
__global__ void gin_segmax_kernel(const float* __restrict__ h,
                                  const int* __restrict__ batch,
                                  float* __restrict__ out, int layer) {
    __shared__ int range[2];
    int g = blockIdx.x;
    if (threadIdx.x < 2) {
        int target = g + (int)threadIdx.x;
        int lo = 0, hi = N_NODES;
        while (lo < hi) {
            int mid = (lo + hi) >> 1;
            if (batch[mid] < target) lo = mid + 1; else hi = mid;
        }
        range[threadIdx.x] = lo;
    }
    __syncthreads();
    int f = threadIdx.x;                        // 0..127
    float m = -__builtin_inff();
    for (int i = range[0]; i < range[1]; ++i)
        m = fmaxf(m, h[i * DIM + f]);
    out[g * (N_LAYERS * DIM) + layer * DIM + f] = m;
}

// ----------------------------------------------------------------- dispatch
extern "C" void kernel_launch(void* const* d_in, const int* in_sizes, int n_in,
                              void* d_out, int out_size, void* d_ws, size_t ws_size,
                              hipStream_t stream) {
    const float* x     = (const float*)d_in[0];
    const int*   ei    = (const int*)d_in[1];
    const int*   esrc  = ei;                // edge_index[0]
    const int*   edst  = ei + N_EDGES;      // edge_index[1]
    const int*   batch = (const int*)d_in[2];
    const float* W1_0  = (const float*)d_in[3];
    const float* b1_0  = (const float*)d_in[4];
    const float* W2_0  = (const float*)d_in[5];
    const float* b2_0  = (const float*)d_in[6];
    const float* W1_r  = (const float*)d_in[7];
    const float* b1_r  = (const float*)d_in[8];
    const float* W2_r  = (const float*)d_in[9];
    const float* b2_r  = (const float*)d_in[10];
    const float* gamma = (const float*)d_in[11];
    const float* beta  = (const float*)d_in[12];
    float* out = (float*)d_out;

    char* ws = (char*)d_ws;
    const size_t per = (size_t)N_NODES * DIM * sizeof(float);
    float* t     = (float*)(ws);              // MLP input (h + agg)
    float* hA    = (float*)(ws + per);        // ping
    float* hB    = (float*)(ws + 2 * per);    // pong
    float* stats = (float*)(ws + 3 * per);    // 512 floats
    float* wpad  = stats + 4 * DIM;           // 80x128 padded layer-0 W1

    // one-shot: pad layer-0 W1 to KP0 rows
    gin_pad_w_kernel<<<(KP0 * DIM + 255) / 256, 256, 0, stream>>>(W1_0, wpad);

    for (int layer = 0; layer < N_LAYERS; ++layer) {
        const float *hin, *b1, *W2, *b2;
        float *hout;
        if (layer == 0) {
            hin = x;
            b1 = b1_0; W2 = W2_0; b2 = b2_0;
            hout = hA;
        } else {
            int r = layer - 1;
            hin = (layer == 1) ? hA : hB;
            b1 = b1_r + r * DIM;
            W2 = W2_r + (size_t)r * DIM * DIM; b2 = b2_r + r * DIM;
            hout = (layer == 1) ? hB : hA;
        }

        if (layer == 0) {
            // t = x ; t[dst] += x[src] ; u(=hB) = relu(t@W1+b1) ; hA = relu(u@W2+b2)
            gin_copy_kernel<<<2048, 256, 0, stream>>>(x, t, N_NODES * IN_DIM);
            gin_edge_scatter77_kernel<<<4096, 256, 0, stream>>>(x, esrc, edst, t);
            gin_gemm_bias_relu_wmma<IN_DIM, KP0><<<N_NODES / 16, 256, 0, stream>>>(t, wpad, b1, hB);
            gin_gemm_bias_relu_wmma<DIM, DIM><<<N_NODES / 16, 256, 0, stream>>>(hB, W2, b2, hout);
        } else {
            const float* W1 = W1_r + (size_t)(layer - 1) * DIM * DIM;
            gin_copy_kernel<<<2048, 256, 0, stream>>>(hin, t, N_NODES * DIM);
            gin_edge_scatter128_kernel<<<4096, 256, 0, stream>>>(hin, esrc, edst, t);
            gin_gemm_bias_relu_wmma<DIM, DIM><<<N_NODES / 16, 256, 0, stream>>>(t, W1, b1, t);   // in-place
            gin_gemm_bias_relu_wmma<DIM, DIM><<<N_NODES / 16, 256, 0, stream>>>(t, W2, b2, hout);
        }

        // BatchNorm (training-mode batch stats)
        gin_zero_kernel<<<1, 256, 0, stream>>>(stats, 2 * DIM);
        gin_bn_stats_kernel<<<(N_NODES + 511) / 512, DIM, 0, stream>>>(hout, stats, 512);
        gin_bn_finalize_kernel<<<1, DIM, 0, stream>>>(stats, gamma + layer * DIM, beta + layer * DIM);
        gin_bn_apply_kernel<<<2048, 256, 0, stream>>>(hout, stats);

        // JK-cat + segment_max, written incrementally per layer
        gin_segmax_kernel<<<N_GRAPHS, DIM, 0, stream>>>(hout, batch, out, layer);
    }
}